// DigitCapsLayer_64862596104608
// MI455X (gfx1250) — compile-verified
//
#include <hip/hip_runtime.h>

#define B_  256
#define D_  10
#define P_  1152
#define O_  16
#define I_  8

typedef __attribute__((ext_vector_type(2))) float v2f;
typedef __attribute__((ext_vector_type(8))) float v8f;

// ---------------------------------------------------------------------------
// Kernel 1: transpose W[1,D,P,O,I] (f32) -> Wt[D,P,I,O] so the WMMA B-operand
// loads are lane-contiguous (16 consecutive o per i-row).
// ---------------------------------------------------------------------------
__global__ void transpose_w_kernel(const float* __restrict__ W,
                                   float* __restrict__ Wt) {
    int idx = blockIdx.x * blockDim.x + threadIdx.x;
    const int total = D_ * P_ * O_ * I_;
    if (idx >= total) return;
    int i  = idx & 7;          // innermost of W
    int o  = (idx >> 3) & 15;
    int dp = idx >> 7;         // d*P + p
    Wt[((size_t)dp * I_ + i) * O_ + o] = W[idx];
}

// ---------------------------------------------------------------------------
// Kernel 2: u_hat[b,d,p,o] = sum_i W[d,p,o,i] * x[b,p,i] via f32 WMMA.
// One wave per (p, 16-batch tile); fully unrolled over d so every Wt load and
// every U store folds into a 24-bit immediate offset off a single base
// address pair (d-stride Wt = 589824B, d-stride U = 73728B, r-stride U =
// 737280B, all < 8.38MB).
//   D = A(16x4) x B(4x16):  M = batch row, N = o, K = i.
//   A layout: lanes 0-15 -> {K0,K1} in V0,V1 ; lanes 16-31 -> {K2,K3}.
//   B layout: V0: lanes0-15=K0 row, lanes16-31=K2 row; V1: K1/K3 rows.
//   D layout: VGPR r: lanes0-15 -> D[M=r,N=lane]; lanes16-31 -> D[M=r+8,...].
// ---------------------------------------------------------------------------
__global__ __launch_bounds__(256) void uhat_wmma_kernel(
        const float* __restrict__ x,   // [B,P,I]
        const float* __restrict__ Wt,  // [D,P,I,O]
        float* __restrict__ U) {       // [B,D,P,O]
    const int lane = threadIdx.x & 31;
    const int unit = blockIdx.x * 8 + (threadIdx.x >> 5); // exact grid, no guard
    const int p  = unit % P_;
    const int b0 = (unit / P_) * 16;

    const int lrow = lane & 15;   // N column / A-row within half
    const int lhi  = lane >> 4;   // 0: K0..1, 1: K2..3
    const int ib   = lhi * 2;

    // A operand from x (reused for all 10 d): row = batch, K = i
    const float* xr = x + ((size_t)(b0 + lrow) * P_ + p) * I_;
    v2f a0, a1;
    a0[0] = xr[ib + 0]; a0[1] = xr[ib + 1];   // K-half 0: i in {0..3}
    a1[0] = xr[ib + 4]; a1[1] = xr[ib + 5];   // K-half 1: i in {4..7}

    // Single base pointers; all strides become immediate offsets.
    const float* wbase = Wt + (size_t)p * (I_ * O_) + ib * O_ + lrow;
    float* ubase = U + ((size_t)(b0 + lhi * 8) * D_ * P_ + p) * O_ + lrow;

    #pragma unroll
    for (int d = 0; d < D_; ++d) {
        const int wd = d * (P_ * I_ * O_);          // 147456 floats per d
        v2f bv0, bv1;
        bv0[0] = wbase[wd +  0];                    // (ib+0)*16
        bv0[1] = wbase[wd + 16];                    // (ib+1)*16
        bv1[0] = wbase[wd + 64];                    // (ib+4)*16
        bv1[1] = wbase[wd + 80];                    // (ib+5)*16

        v8f c = {};
        c = __builtin_amdgcn_wmma_f32_16x16x4_f32(false, a0, false, bv0,
                                                  (short)0, c, false, false);
        c = __builtin_amdgcn_wmma_f32_16x16x4_f32(false, a1, false, bv1,
                                                  (short)0, c, false, false);

        #pragma unroll
        for (int r = 0; r < 8; ++r)
            ubase[(size_t)r * (D_ * P_ * O_) + (size_t)d * (P_ * O_)] = c[r];
    }
}

// ---------------------------------------------------------------------------
// Kernel 3: fused dynamic routing, one 256-thread block per batch element.
// Logits/couplings in LDS; u_hat re-read from L2-resident workspace.
// ---------------------------------------------------------------------------
__global__ __launch_bounds__(256) void routing_kernel(
        const float* __restrict__ U,   // [B,D,P,O]
        float* __restrict__ out) {     // [B,D,O]
    __shared__ float blog[D_][P_];
    __shared__ float cmat[D_][P_];
    __shared__ float red[D_][8][O_];
    __shared__ float sv[D_][O_];
    __shared__ float vv[D_][O_];

    const int tid  = threadIdx.x;
    const int lane = tid & 31;
    const int wv   = tid >> 5;
    const int b    = blockIdx.x;
    const float* Ub = U + (size_t)b * D_ * P_ * O_;

    for (int idx = tid; idx < D_ * P_; idx += 256) (&blog[0][0])[idx] = 0.f;
    __syncthreads();

    for (int it = 0; it < 3; ++it) {
        // ---- pass A: c = softmax_d(blog) ----
        for (int p = tid; p < P_; p += 256) {
            float lg[D_];
            float mx = -1e30f;
            #pragma unroll
            for (int d = 0; d < D_; ++d) { lg[d] = blog[d][p]; mx = fmaxf(mx, lg[d]); }
            float den = 0.f;
            #pragma unroll
            for (int d = 0; d < D_; ++d) { lg[d] = __expf(lg[d] - mx); den += lg[d]; }
            float inv = 1.0f / den;
            #pragma unroll
            for (int d = 0; d < D_; ++d) cmat[d][p] = lg[d] * inv;
        }
        __syncthreads();

        // ---- pass B: s[d][o] = sum_p c[d][p] * u[d][p][o] ----
        for (int d = 0; d < D_; ++d) {
            float acc[O_];
            #pragma unroll
            for (int o = 0; o < O_; ++o) acc[o] = 0.f;
            for (int p = tid; p < P_; p += 256) {
                float cv = cmat[d][p];
                const float4* up =
                    (const float4*)(Ub + ((size_t)d * P_ + p) * O_);
                float4 u0 = up[0], u1 = up[1], u2 = up[2], u3 = up[3];
                acc[0]  += cv * u0.x; acc[1]  += cv * u0.y;
                acc[2]  += cv * u0.z; acc[3]  += cv * u0.w;
                acc[4]  += cv * u1.x; acc[5]  += cv * u1.y;
                acc[6]  += cv * u1.z; acc[7]  += cv * u1.w;
                acc[8]  += cv * u2.x; acc[9]  += cv * u2.y;
                acc[10] += cv * u2.z; acc[11] += cv * u2.w;
                acc[12] += cv * u3.x; acc[13] += cv * u3.y;
                acc[14] += cv * u3.z; acc[15] += cv * u3.w;
            }
            #pragma unroll
            for (int m = 16; m >= 1; m >>= 1) {
                #pragma unroll
                for (int o = 0; o < O_; ++o) acc[o] += __shfl_xor(acc[o], m, 32);
            }
            if (lane == 0) {
                #pragma unroll
                for (int o = 0; o < O_; ++o) red[d][wv][o] = acc[o];
            }
        }
        __syncthreads();
        if (tid < D_ * O_) {
            int d = tid >> 4, o = tid & 15;
            float s = 0.f;
            #pragma unroll
            for (int w = 0; w < 8; ++w) s += red[d][w][o];
            sv[d][o] = s;
        }
        __syncthreads();
        // ---- squash per d ----
        if (tid < D_) {
            float sq = 0.f;
            #pragma unroll
            for (int o = 0; o < O_; ++o) { float t = sv[tid][o]; sq += t * t; }
            float scale = (sq / (1.0f + sq)) * rsqrtf(sq + 1e-7f);
            #pragma unroll
            for (int o = 0; o < O_; ++o) vv[tid][o] = sv[tid][o] * scale;
        }
        __syncthreads();

        // ---- pass C: blog[d][p] += sum_o u[d][p][o] * v[d][o] ----
        if (it < 2) {
            for (int p = tid; p < P_; p += 256) {
                #pragma unroll
                for (int d = 0; d < D_; ++d) {
                    const float4* up =
                        (const float4*)(Ub + ((size_t)d * P_ + p) * O_);
                    float4 u0 = up[0], u1 = up[1], u2 = up[2], u3 = up[3];
                    float dot =
                        u0.x * vv[d][0]  + u0.y * vv[d][1]  +
                        u0.z * vv[d][2]  + u0.w * vv[d][3]  +
                        u1.x * vv[d][4]  + u1.y * vv[d][5]  +
                        u1.z * vv[d][6]  + u1.w * vv[d][7]  +
                        u2.x * vv[d][8]  + u2.y * vv[d][9]  +
                        u2.z * vv[d][10] + u2.w * vv[d][11] +
                        u3.x * vv[d][12] + u3.y * vv[d][13] +
                        u3.z * vv[d][14] + u3.w * vv[d][15];
                    blog[d][p] += dot;
                }
            }
            __syncthreads();
        }
    }

    if (tid < D_ * O_)
        out[(size_t)b * (D_ * O_) + tid] = vv[tid >> 4][tid & 15];
}

// ---------------------------------------------------------------------------
extern "C" void kernel_launch(void* const* d_in, const int* in_sizes, int n_in,
                              void* d_out, int out_size, void* d_ws, size_t ws_size,
                              hipStream_t stream) {
    (void)in_sizes; (void)n_in; (void)out_size; (void)ws_size;
    const float* x = (const float*)d_in[0];   // [B,P,I] f32
    const float* W = (const float*)d_in[1];   // [1,D,P,O,I] f32
    float* out = (float*)d_out;               // [B,D,O] f32

    const size_t wt_bytes = (size_t)D_ * P_ * O_ * I_ * sizeof(float); // 5.9MB, 256B-mult
    float* Wt = (float*)d_ws;
    float* U  = (float*)((char*)d_ws + wt_bytes);                      // 189MB

    const int wtotal = D_ * P_ * O_ * I_;
    transpose_w_kernel<<<(wtotal + 255) / 256, 256, 0, stream>>>(W, Wt);

    const int units = P_ * (B_ / 16);          // 18432 waves, 8 waves/block
    uhat_wmma_kernel<<<units / 8, 256, 0, stream>>>(x, Wt, U);

    routing_kernel<<<B_, 256, 0, stream>>>(U, out);
}